// DTDFIICell_37641093382424
// MI455X (gfx1250) — compile-verified
//
#include <hip/hip_runtime.h>

// ---------------------------------------------------------------------------
// DF-II transposed biquad scan, chunk-parallel, LDS-tiled, TDM/async-pipelined.
//   y    = x_t*b0 + v0
//   v0'  = x_t*b1 + v1 - y*a1
//   v1'  = x_t*b2       - y*a2
// State recurrence is linear: s' = A s + Bx with A = [[-a1,1],[-a2,0]].
// Pass 1: per-(row,chunk) scan with zero entering state -> p (workspace)
// Pass 2: chain entering states with M = A^chunkLen (repeated squaring)
// Pass 3: true scan per chunk from correct entering state -> y, v_final
// ---------------------------------------------------------------------------

#define BIQ_TILE 32            // timesteps per LDS tile
#define ROWS     256           // rows per block (== blockDim.x, all lanes scan)
#define LSTR     34            // padded LDS row stride (floats): even (8B align
                               // for b64), and (34*L + t) % 64 is conflict-free
#define NCHUNKS  16

// ---- CDNA5 feature probes --------------------------------------------------
#if defined(__has_builtin)
#if __has_builtin(__builtin_amdgcn_global_load_async_to_lds_b64) && \
    __has_builtin(__builtin_amdgcn_s_wait_asynccnt)
#define BIQ_ASYNC 1
#endif
#if __has_builtin(__builtin_amdgcn_tensor_load_to_lds) && \
    __has_builtin(__builtin_amdgcn_s_wait_tensorcnt)
#define BIQ_TDM 1
#endif
#endif
#ifndef BIQ_ASYNC
#define BIQ_ASYNC 0
#endif
#ifndef BIQ_TDM
#define BIQ_TDM 0
#endif

typedef int      biq_v2i __attribute__((ext_vector_type(2)));
typedef int      biq_v4i __attribute__((ext_vector_type(4)));
typedef int      biq_v8i __attribute__((ext_vector_type(8)));
typedef unsigned biq_v4u __attribute__((ext_vector_type(4)));

static_assert(BIQ_TILE == 32 && ROWS == 256 && LSTR == 34,
              "TDM descriptor pad fields assume 32x256 tile, stride 34");

__device__ __forceinline__ void lds_copy_b64(const float* g, float* l) {
#if BIQ_ASYNC
  __builtin_amdgcn_global_load_async_to_lds_b64(
      (__attribute__((address_space(1))) biq_v2i*)g,
      (__attribute__((address_space(3))) biq_v2i*)l,
      /*offset=*/0, /*cpol=*/0);
#else
  *(float2*)l = *(const float2*)g;   // sync fallback: global->VGPR->LDS
#endif
}

// Issue the copy of one 256-row x 32-float tile (row stride T) into LDS at
// ldsBase with padded row stride LSTR. TDM path: single descriptor-driven DMA
// issued by wave 0; hardware applies LDS padding (pad_interval=32 DW,
// pad_amount=2 DW -> stride 34 floats). Fallback: per-wave async b64 copies.
__device__ __forceinline__ void tile_issue(const float* __restrict__ x,
                                           int rowBase, int t0,
                                           float* ldsBase, int T,
                                           int tid, int lane16, int rgrp) {
#if BIQ_TDM
  if (tid < 32) {  // wave 0 only: TDM is a wave-level DMA op
    const float* g = x + (size_t)rowBase * T + t0;
    const unsigned long long ga = (unsigned long long)(size_t)g;
    const unsigned ldsAddr =
        (unsigned)(size_t)(__attribute__((address_space(3))) float*)ldsBase;

    biq_v4u g0;
    g0[0] = 1u;                                   // count=1, no gather
    g0[1] = ldsAddr;                              // lds_addr [63:32]
    g0[2] = (unsigned)ga;                         // global_addr [95:64]
    g0[3] = (unsigned)((ga >> 32) & 0x01FFFFFFu)  // global_addr [120:96]
            | (2u << 30);                         // type=2 ("image")

    biq_v8i g1;
    g1[0] = (int)((2u << 16)       // data_size = 4 bytes
                  | (1u << 20)     // pad_enable
                  | (4u << 22)     // pad_interval: 32 DWORDs
                  | (1u << 25));   // pad_amount: 2 DWORDs
    g1[1] = (int)((unsigned)BIQ_TILE << 16);  // tensor_dim0[15:0] = 32
    g1[2] = (int)((unsigned)ROWS << 16);      // tensor_dim1[15:0] = 256
    g1[3] = (int)((unsigned)BIQ_TILE << 16);  // tile_dim0 = 32
    g1[4] = ROWS;                             // tile_dim1 = 256
    g1[5] = T;                                // tensor_dim0_stride = T
    g1[6] = 0;
    g1[7] = 0;

    const biq_v4i z4 = {0, 0, 0, 0};
#if __clang_major__ >= 23
    const biq_v8i z8 = {0, 0, 0, 0, 0, 0, 0, 0};
    __builtin_amdgcn_tensor_load_to_lds(g0, g1, z4, z4, z8, 0);
#else
    __builtin_amdgcn_tensor_load_to_lds(g0, g1, z4, z4, 0);
#endif
  }
#else
  for (int j = 0; j < 16; ++j) {
    const int r = rgrp + 16 * j;
    lds_copy_b64(x + (size_t)(rowBase + r) * T + t0 + 2 * lane16,
                 ldsBase + r * LSTR + 2 * lane16);
  }
  (void)tid;
#endif
}

__device__ __forceinline__ void wait_tile(bool more_inflight, int tid) {
#if BIQ_TDM
  if (tid < 32) {  // only the issuing wave's TENSORcnt is meaningful
    if (more_inflight) __builtin_amdgcn_s_wait_tensorcnt(1);
    else               __builtin_amdgcn_s_wait_tensorcnt(0);
  }
#elif BIQ_ASYNC
  // 16 b64 async copies per wave per tile, in-order completion: waiting to
  // <=16 outstanding means the previous tile has fully landed.
  if (more_inflight) __builtin_amdgcn_s_wait_asynccnt(16);
  else               __builtin_amdgcn_s_wait_asynccnt(0);
#else
  (void)more_inflight;
  (void)tid;
#endif
}

// ---------------------------------------------------------------------------
// Tiled scan kernel. WRITE_Y=false: pass 1 (zero init, emit chunk states).
//                    WRITE_Y=true : pass 3 (sIn init, emit y and v_final).
// grid = (B/ROWS, nChunks), block = ROWS threads (8 waves).
// ---------------------------------------------------------------------------
template <bool WRITE_Y>
__global__ void __launch_bounds__(ROWS)
biquad_scan_kernel(const float* __restrict__ x,
                   const float* __restrict__ gb0, const float* __restrict__ gb1,
                   const float* __restrict__ gb2, const float* __restrict__ ga1,
                   const float* __restrict__ ga2,
                   const float2* __restrict__ sIn,   // null => zero init
                   float* __restrict__ yOut,         // used when WRITE_Y
                   float2* __restrict__ stateOut,    // !WRITE_Y: p[row][chunk]
                                                     //  WRITE_Y: v_final[row]
                   int T, int nChunks) {
  __shared__ float xs[2][ROWS * LSTR];
  __shared__ float ys[WRITE_Y ? (ROWS * LSTR) : 2];

  const int tid     = threadIdx.x;
  const int rowBase = blockIdx.x * ROWS;
  const int c       = blockIdx.y;
  const int CL      = T / nChunks;
  const int tOff    = c * CL;
  const int NT      = CL / BIQ_TILE;

  const float b0 = *gb0, b1 = *gb1, b2 = *gb2, a1 = *ga1, a2 = *ga2;

  float v0s = 0.f, v1s = 0.f;
  if (sIn != nullptr) {
    float2 s = sIn[(size_t)(rowBase + tid) * nChunks + c];
    v0s = s.x;
    v1s = s.y;
  }

  // Fallback copy mapping: lanes 0..15 cover one row's 32 floats (128B,
  // coalesced), lanes 16..31 the next row; 16 iterations cover 256 rows.
  const int lane16 = tid & 15;
  const int rgrp   = tid >> 4;

  // Preload tile 0.
  tile_issue(x, rowBase, tOff, &xs[0][0], T, tid, lane16, rgrp);

  int buf = 0;
  for (int tile = 0; tile < NT; ++tile) {
    const bool have_next = (tile + 1 < NT);
    if (have_next) {
      tile_issue(x, rowBase, tOff + (tile + 1) * BIQ_TILE, &xs[buf ^ 1][0], T,
                 tid, lane16, rgrp);
    }
    wait_tile(have_next, tid);  // issuing wave: previous tile landed in LDS
    __syncthreads();            // propagate to all waves

    // Serial scan: every thread owns one row. LDS reads are conflict-free
    // ((34*lane + t) mod 64 distinct across a wave).
    {
      const float* xr = &xs[buf][tid * LSTR];
      float* yr = WRITE_Y ? &ys[tid * LSTR] : nullptr;
#pragma unroll
      for (int t = 0; t < BIQ_TILE; ++t) {
        const float xt  = xr[t];
        const float y   = fmaf(xt, b0, v0s);
        const float nv0 = fmaf(xt, b1, v1s) - y * a1;
        const float nv1 = xt * b2 - y * a2;
        v0s = nv0;
        v1s = nv1;
        if (WRITE_Y) yr[t] = y;
      }
    }

    if (WRITE_Y) {
      __syncthreads();      // y tile complete in LDS
      const int t0 = tOff + tile * BIQ_TILE;
      for (int j = 0; j < 16; ++j) {
        const int r = rgrp + 16 * j;
        const float2 v = *(const float2*)&ys[r * LSTR + 2 * lane16];
        *(float2*)(yOut + (size_t)(rowBase + r) * T + t0 + 2 * lane16) = v;
      }
    }
    __syncthreads();        // xs[buf] / ys safe to overwrite next iteration
    buf ^= 1;
  }

  if (WRITE_Y) {
    if (c == nChunks - 1 && stateOut != nullptr)
      stateOut[rowBase + tid] = make_float2(v0s, v1s);  // exact v_final
  } else {
    stateOut[(size_t)(rowBase + tid) * nChunks + c] = make_float2(v0s, v1s);
  }
}

// ---------------------------------------------------------------------------
// Pass 2: chain entering states. M = A^CL via repeated squaring (log2 steps);
// s_in(c) = M * s_in(c-1) + p(c-1). One thread per row; trivially cheap.
// ---------------------------------------------------------------------------
__global__ void __launch_bounds__(256)
biquad_combine_kernel(const float2* __restrict__ v0,
                      const float2* __restrict__ p,
                      float2* __restrict__ sIn,
                      const float* __restrict__ ga1,
                      const float* __restrict__ ga2,
                      int CL, int nChunks, int B) {
  const int r = blockIdx.x * blockDim.x + threadIdx.x;
  if (r >= B) return;
  const float a1 = *ga1, a2 = *ga2;
  // A = [[-a1, 1], [-a2, 0]]
  float m00 = -a1, m01 = 1.f, m10 = -a2, m11 = 0.f;
  float M00 = 1.f, M01 = 0.f, M10 = 0.f, M11 = 1.f;
  int e = CL;
  while (e) {
    if (e & 1) {
      const float t00 = M00 * m00 + M01 * m10;
      const float t01 = M00 * m01 + M01 * m11;
      const float t10 = M10 * m00 + M11 * m10;
      const float t11 = M10 * m01 + M11 * m11;
      M00 = t00; M01 = t01; M10 = t10; M11 = t11;
    }
    e >>= 1;
    if (e) {
      const float t00 = m00 * m00 + m01 * m10;
      const float t01 = m00 * m01 + m01 * m11;
      const float t10 = m10 * m00 + m11 * m10;
      const float t11 = m10 * m01 + m11 * m11;
      m00 = t00; m01 = t01; m10 = t10; m11 = t11;
    }
  }
  float sx = v0[r].x, sy = v0[r].y;
  for (int c = 0; c < nChunks; ++c) {
    sIn[(size_t)r * nChunks + c] = make_float2(sx, sy);
    const float2 pc = p[(size_t)r * nChunks + c];
    const float nx = M00 * sx + M01 * sy + pc.x;
    const float ny = M10 * sx + M11 * sy + pc.y;
    sx = nx; sy = ny;
  }
}

// ---------------------------------------------------------------------------
// Safety-net kernel for shapes the tiled path can't handle.
// ---------------------------------------------------------------------------
__global__ void __launch_bounds__(256)
biquad_naive_kernel(const float* __restrict__ x, const float2* __restrict__ v0,
                    const float* __restrict__ gb0, const float* __restrict__ gb1,
                    const float* __restrict__ gb2, const float* __restrict__ ga1,
                    const float* __restrict__ ga2, float* __restrict__ y,
                    float2* __restrict__ vfin, int B, int T) {
  const int r = blockIdx.x * blockDim.x + threadIdx.x;
  if (r >= B) return;
  const float b0 = *gb0, b1 = *gb1, b2 = *gb2, a1 = *ga1, a2 = *ga2;
  float v0s = v0[r].x, v1s = v0[r].y;
  const float* xr = x + (size_t)r * T;
  float* yr = y + (size_t)r * T;
  for (int t = 0; t < T; ++t) {
    const float xt  = xr[t];
    const float yy  = fmaf(xt, b0, v0s);
    const float nv0 = fmaf(xt, b1, v1s) - yy * a1;
    const float nv1 = xt * b2 - yy * a2;
    v0s = nv0;
    v1s = nv1;
    yr[t] = yy;
  }
  vfin[r] = make_float2(v0s, v1s);
}

// ---------------------------------------------------------------------------
extern "C" void kernel_launch(void* const* d_in, const int* in_sizes, int n_in,
                              void* d_out, int out_size, void* d_ws,
                              size_t ws_size, hipStream_t stream) {
  const float*  x  = (const float*)d_in[0];
  const float2* v0 = (const float2*)d_in[1];
  const float* b0 = (const float*)d_in[2];
  const float* b1 = (const float*)d_in[3];
  const float* b2 = (const float*)d_in[4];
  const float* a1 = (const float*)d_in[5];
  const float* a2 = (const float*)d_in[6];

  const int B = in_sizes[1] / 2;
  const int T = in_sizes[0] / B;

  float*  y    = (float*)d_out;
  float2* vfin = (float2*)(y + (size_t)B * T);

  const size_t need = (size_t)B * NCHUNKS * sizeof(float2) * 2;  // p + sIn
  const bool tiled_ok = (B % ROWS == 0) && (T % BIQ_TILE == 0);

  if (!tiled_ok) {
    biquad_naive_kernel<<<(B + 255) / 256, 256, 0, stream>>>(
        x, v0, b0, b1, b2, a1, a2, y, vfin, B, T);
    return;
  }

  int nChunks = NCHUNKS;
  if (ws_size < need || (T % (NCHUNKS * BIQ_TILE)) != 0) nChunks = 1;

  if (nChunks > 1) {
    float2* p   = (float2*)d_ws;
    float2* sIn = p + (size_t)B * nChunks;
    dim3 grid(B / ROWS, nChunks);
    // Pass 1: chunk-local states with zero entering state.
    biquad_scan_kernel<false><<<grid, ROWS, 0, stream>>>(
        x, b0, b1, b2, a1, a2, /*sIn=*/nullptr, /*yOut=*/nullptr,
        /*stateOut=*/p, T, nChunks);
    // Pass 2: chain entering states across chunks.
    biquad_combine_kernel<<<(B + 255) / 256, 256, 0, stream>>>(
        v0, p, sIn, a1, a2, T / nChunks, nChunks, B);
    // Pass 3: true scan per chunk, emit y and exact v_final.
    biquad_scan_kernel<true><<<grid, ROWS, 0, stream>>>(
        x, b0, b1, b2, a1, a2, sIn, y, vfin, T, nChunks);
  } else {
    dim3 grid(B / ROWS, 1);
    biquad_scan_kernel<true><<<grid, ROWS, 0, stream>>>(
        x, b0, b1, b2, a1, a2, v0, y, vfin, T, 1);
  }
}